// ResidualBlock_257698038141
// MI455X (gfx1250) — compile-verified
//
#include <hip/hip_runtime.h>

// Problem constants (from the reference)
#define N_VOX  400000
#define CCH    64
#define KOFF   27
#define MPAIR  150000
#define BN_EPS 1e-5f

typedef float v2f __attribute__((ext_vector_type(2)));
typedef float v8f __attribute__((ext_vector_type(8)));
typedef int   v4i __attribute__((ext_vector_type(4)));

// ---------------------------------------------------------------------------
// Zero-fill workspace (accumulation targets for the scatter-add convs).
// ---------------------------------------------------------------------------
__global__ __launch_bounds__(256) void zero_f4(float4* __restrict__ p, long n4) {
    long i = (long)blockIdx.x * blockDim.x + threadIdx.x;
    long stride = (long)gridDim.x * blockDim.x;
    float4 z = make_float4(0.f, 0.f, 0.f, 0.f);
    for (; i < n4; i += stride) p[i] = z;
}

// ---------------------------------------------------------------------------
// Sparse conv: for offset k = blockIdx.y, pairs chunk = blockIdx.x.
// out[nbr_out[k][m]] += x[nbr_in[k][m]] @ W[k]   (M x 64 @ 64 x 64)
// One wave handles 16 pairs -> 16x64 fp32 tile via V_WMMA_F32_16X16X4_F32.
// W_k is staged into LDS with the CDNA5 async global->LDS path (ASYNCcnt).
// ---------------------------------------------------------------------------
__global__ __launch_bounds__(256) void spconv_wmma_f32(
    const float* __restrict__ xin,      // [N, 64]
    const float* __restrict__ W,        // [K, 64, 64] (cin, cout)
    const int*   __restrict__ nbr_in,   // [K, M]
    const int*   __restrict__ nbr_out,  // [K, M]
    float*       __restrict__ outbuf)   // [N, 64] accumulated atomically
{
    __shared__ float lw[64 * 64];       // W_k, row-major, 16 KB

    const int k = blockIdx.y;
    const float* __restrict__ Wk = W + (size_t)k * 64 * 64;

#if __has_builtin(__builtin_amdgcn_global_load_async_to_lds_b128)
    // 1024 x 16B chunks, 4 per thread: global_load_async_to_lds_b128
    for (int i = threadIdx.x; i < 64 * 16; i += 256) {
        __builtin_amdgcn_global_load_async_to_lds_b128(
            (v4i*)(Wk + i * 4), (v4i*)(lw + i * 4), 0, 0);
    }
  #if __has_builtin(__builtin_amdgcn_s_wait_asynccnt)
    __builtin_amdgcn_s_wait_asynccnt(0);
  #else
    asm volatile("s_wait_asynccnt 0x0" ::: "memory");
  #endif
#else
    for (int i = threadIdx.x; i < 64 * 16; i += 256) {
        ((float4*)lw)[i] = ((const float4*)Wk)[i];
    }
#endif
    __syncthreads();

    const int lane  = threadIdx.x & 31;
    const int wv    = threadIdx.x >> 5;
    const int pair0 = (blockIdx.x * 8 + wv) * 16;
    if (pair0 + 16 > MPAIR) return;     // wave-uniform: EXEC stays all-ones

    const int mlo = lane & 15;          // row (A) / column (B,C,D) within tile
    const int hi  = lane >> 4;          // half-wave selects K sub-pair

    const int* __restrict__ nin  = nbr_in  + (size_t)k * MPAIR + pair0;
    const int* __restrict__ nout = nbr_out + (size_t)k * MPAIR + pair0;

    // A-matrix layout (f32 16x4): lane(0-15)=M, VGPR0/1 = K {0,1} (lo half)
    // or K {2,3} (hi half)  -> one contiguous float2 per lane per K-step.
    const float* __restrict__ xrow = xin + (size_t)nin[mlo] * CCH;

    v8f a0 = {}, a1 = {}, a2 = {}, a3 = {};   // 4 x (16x16) column groups

#pragma unroll
    for (int c0 = 0; c0 < CCH; c0 += 4) {
        v2f av = *(const v2f*)(xrow + c0 + 2 * hi);

        // B-matrix (4x16): VGPR0/1 hold rows {0,1} (lo half) / {2,3} (hi half),
        // column = lane&15.  b.x = W[c0+2hi][n], b.y = W[c0+2hi+1][n].
        const float* __restrict__ lp = lw + (c0 + 2 * hi) * 64 + mlo;
        v2f b0; b0.x = lp[0];  b0.y = lp[64];        // cols  0..15
        v2f b1; b1.x = lp[16]; b1.y = lp[80];        // cols 16..31
        v2f b2; b2.x = lp[32]; b2.y = lp[96];        // cols 32..47
        v2f b3; b3.x = lp[48]; b3.y = lp[112];       // cols 48..63

        a0 = __builtin_amdgcn_wmma_f32_16x16x4_f32(false, av, false, b0, (short)0, a0, false, false);
        a1 = __builtin_amdgcn_wmma_f32_16x16x4_f32(false, av, false, b1, (short)0, a1, false, false);
        a2 = __builtin_amdgcn_wmma_f32_16x16x4_f32(false, av, false, b2, (short)0, a2, false, false);
        a3 = __builtin_amdgcn_wmma_f32_16x16x4_f32(false, av, false, b3, (short)0, a3, false, false);
    }

    // C/D layout: VGPR j -> M = j + 8*hi, N = lane&15. Scatter with f32 atomics.
#pragma unroll
    for (int j = 0; j < 8; ++j) {
        const int m = j + 8 * hi;
        float* __restrict__ op = outbuf + (size_t)nout[m] * CCH + mlo;
        atomicAdd(op +  0, a0[j]);
        atomicAdd(op + 16, a1[j]);
        atomicAdd(op + 32, a2[j]);
        atomicAdd(op + 48, a3[j]);
    }
}

// ---------------------------------------------------------------------------
// Fused eval-mode BN + ReLU, in place:  h = relu(h*scale + shift)
// scale = gamma*rsqrt(var+eps), shift = beta - mean*scale
// ---------------------------------------------------------------------------
__global__ __launch_bounds__(256) void bn_relu_inplace(
    float* __restrict__ h,
    const float* __restrict__ gamma, const float* __restrict__ beta,
    const float* __restrict__ mean,  const float* __restrict__ var)
{
    __shared__ float sc[64], sb[64];
    if (threadIdx.x < 64) {
        float s = gamma[threadIdx.x] * rsqrtf(var[threadIdx.x] + BN_EPS);
        sc[threadIdx.x] = s;
        sb[threadIdx.x] = beta[threadIdx.x] - mean[threadIdx.x] * s;
    }
    __syncthreads();

    float4* __restrict__ h4 = (float4*)h;
    const long n4 = (long)N_VOX * (CCH / 4);
    long i = (long)blockIdx.x * blockDim.x + threadIdx.x;
    const long stride = (long)gridDim.x * blockDim.x;
    for (; i < n4; i += stride) {
        const int cb = (int)(i & 15) << 2;   // channel base of this float4
        float4 v = h4[i];
        v.x = fmaxf(v.x * sc[cb + 0] + sb[cb + 0], 0.f);
        v.y = fmaxf(v.y * sc[cb + 1] + sb[cb + 1], 0.f);
        v.z = fmaxf(v.z * sc[cb + 2] + sb[cb + 2], 0.f);
        v.w = fmaxf(v.w * sc[cb + 3] + sb[cb + 3], 0.f);
        h4[i] = v;
    }
}

// ---------------------------------------------------------------------------
// Final fuse: out = relu(bn2(y_pre) + x)
// ---------------------------------------------------------------------------
__global__ __launch_bounds__(256) void bn_residual_relu(
    const float* __restrict__ ypre, const float* __restrict__ xin,
    const float* __restrict__ gamma, const float* __restrict__ beta,
    const float* __restrict__ mean,  const float* __restrict__ var,
    float* __restrict__ out)
{
    __shared__ float sc[64], sb[64];
    if (threadIdx.x < 64) {
        float s = gamma[threadIdx.x] * rsqrtf(var[threadIdx.x] + BN_EPS);
        sc[threadIdx.x] = s;
        sb[threadIdx.x] = beta[threadIdx.x] - mean[threadIdx.x] * s;
    }
    __syncthreads();

    const float4* __restrict__ y4 = (const float4*)ypre;
    const float4* __restrict__ x4 = (const float4*)xin;
    float4* __restrict__ o4 = (float4*)out;
    const long n4 = (long)N_VOX * (CCH / 4);
    long i = (long)blockIdx.x * blockDim.x + threadIdx.x;
    const long stride = (long)gridDim.x * blockDim.x;
    for (; i < n4; i += stride) {
        const int cb = (int)(i & 15) << 2;
        float4 v = y4[i];
        float4 r = x4[i];
        v.x = fmaxf(v.x * sc[cb + 0] + sb[cb + 0] + r.x, 0.f);
        v.y = fmaxf(v.y * sc[cb + 1] + sb[cb + 1] + r.y, 0.f);
        v.z = fmaxf(v.z * sc[cb + 2] + sb[cb + 2] + r.z, 0.f);
        v.w = fmaxf(v.w * sc[cb + 3] + sb[cb + 3] + r.w, 0.f);
        o4[i] = v;
    }
}

// ---------------------------------------------------------------------------
extern "C" void kernel_launch(void* const* d_in, const int* in_sizes, int n_in,
                              void* d_out, int out_size, void* d_ws, size_t ws_size,
                              hipStream_t stream) {
    const float* x      = (const float*)d_in[0];
    const int*   nbrin  = (const int*)  d_in[1];
    const int*   nbrout = (const int*)  d_in[2];
    const float* W1     = (const float*)d_in[3];
    const float* gamma1 = (const float*)d_in[4];
    const float* beta1  = (const float*)d_in[5];
    const float* mean1  = (const float*)d_in[6];
    const float* var1   = (const float*)d_in[7];
    const float* W2     = (const float*)d_in[8];
    const float* gamma2 = (const float*)d_in[9];
    const float* beta2  = (const float*)d_in[10];
    const float* mean2  = (const float*)d_in[11];
    const float* var2   = (const float*)d_in[12];

    float* hbuf = (float*)d_ws;                       // [N,64] conv1 accum / h
    float* ybuf = hbuf + (size_t)N_VOX * CCH;         // [N,64] conv2 accum

    // Zero both accumulation buffers (2 * N * 64 floats = N*32 float4)
    zero_f4<<<8192, 256, 0, stream>>>((float4*)d_ws, (long)N_VOX * 32);

    const int waves  = MPAIR / 16;                    // 9375 (M % 16 == 0)
    const int gx     = (waves + 7) / 8;               // 8 waves per block
    dim3 cgrid(gx, KOFF);

    spconv_wmma_f32<<<cgrid, 256, 0, stream>>>(x, W1, nbrin, nbrout, hbuf);
    bn_relu_inplace<<<4096, 256, 0, stream>>>(hbuf, gamma1, beta1, mean1, var1);
    spconv_wmma_f32<<<cgrid, 256, 0, stream>>>(hbuf, W2, nbrin, nbrout, ybuf);
    bn_residual_relu<<<4096, 256, 0, stream>>>(ybuf, x, gamma2, beta2, mean2, var2,
                                               (float*)d_out);
}